// SentenceGNN_32976758899203
// MI455X (gfx1250) — compile-verified
//
#include <hip/hip_runtime.h>

typedef __attribute__((ext_vector_type(2))) float v2f;
typedef __attribute__((ext_vector_type(8))) float v8f;
typedef __attribute__((ext_vector_type(4))) int   v4i;

#define N_NODES 100000
#define E_EDGES 3200000
#define IN_DIM  768
#define HID     128
#define NCLS    7

#define GLOBAL_AS __attribute__((address_space(1)))
#define LDS_AS    __attribute__((address_space(3)))

// ---- CDNA5 async global->LDS copy (16B per lane), with safe fallback ----
__device__ __forceinline__ void async_copy16(const float4* g, float4* l) {
#if __has_builtin(__builtin_amdgcn_global_load_async_to_lds_b128)
    __builtin_amdgcn_global_load_async_to_lds_b128(
        (GLOBAL_AS v4i*)g, (LDS_AS v4i*)l, 0, 0);
#elif __has_builtin(__builtin_amdgcn_global_load_async_to_lds_b32)
    const float* gf = (const float*)g;
    float* lf = (float*)l;
    __builtin_amdgcn_global_load_async_to_lds_b32((GLOBAL_AS int*)(gf + 0), (LDS_AS int*)(lf + 0), 0, 0);
    __builtin_amdgcn_global_load_async_to_lds_b32((GLOBAL_AS int*)(gf + 1), (LDS_AS int*)(lf + 1), 0, 0);
    __builtin_amdgcn_global_load_async_to_lds_b32((GLOBAL_AS int*)(gf + 2), (LDS_AS int*)(lf + 2), 0, 0);
    __builtin_amdgcn_global_load_async_to_lds_b32((GLOBAL_AS int*)(gf + 3), (LDS_AS int*)(lf + 3), 0, 0);
#else
    *l = *g;  // synchronous VGPR staging fallback
#endif
}

__device__ __forceinline__ void wait_async_all() {
#if __has_builtin(__builtin_amdgcn_s_wait_asynccnt)
    __builtin_amdgcn_s_wait_asynccnt(0);
#else
    asm volatile("s_wait_asynccnt 0" ::: "memory");
#endif
}

// ---------------- degree / normalization ----------------
__global__ void init_deg_k(float* __restrict__ deg, int n) {
    int i = blockIdx.x * blockDim.x + threadIdx.x;
    if (i < n) deg[i] = 1.0f;  // self loop
}

__global__ void count_deg_k(const int* __restrict__ dst, float* __restrict__ deg, int e) {
    int i = blockIdx.x * blockDim.x + threadIdx.x;
    if (i < e) atomicAdd(&deg[dst[i]], 1.0f);
}

__global__ void dinv_k(float* __restrict__ deg, int n) {
    int i = blockIdx.x * blockDim.x + threadIdx.x;
    if (i < n) deg[i] = rsqrtf(deg[i]);  // deg >= 1 always
}

// ---------------- WMMA f32 GEMM: C[M,NC] = A[M,K] * B[K,NC] ----------------
// Block: 16 rows x NC cols, NC/16 waves (wave32). Double-buffered async LDS staging.
template <int K, int NC>
__global__ void gemm16_wmma_k(const float* __restrict__ A,
                              const float* __restrict__ B,
                              float* __restrict__ C) {
    __shared__ __align__(16) float sA[2][16 * 32];
    __shared__ __align__(16) float sB[2][32 * NC];

    constexpr int NC4 = NC / 4;          // float4s per B row
    const int tid   = threadIdx.x;
    const int nthr  = blockDim.x;
    const int wave  = tid >> 5;
    const int lane  = tid & 31;
    const int half  = lane >> 4;         // 0: lanes 0-15, 1: lanes 16-31
    const int l15   = lane & 15;
    const int mbase = blockIdx.x * 16;
    const int nbase = wave * 16;

    v8f acc = {};

    // ---- prologue: stage chunk 0 into buffer 0 ----
    for (int f = tid; f < 128; f += nthr) {           // A tile: 16x32 = 128 float4
        int r = f >> 3, c = f & 7;
        async_copy16((const float4*)(A + (size_t)(mbase + r) * K + 4 * c),
                     (float4*)(&sA[0][r * 32 + c * 4]));
    }
    for (int f = tid; f < 8 * NC; f += nthr) {        // B tile: 32xNC = 8*NC float4
        int r = f / NC4, c = f % NC4;
        async_copy16((const float4*)(B + (size_t)r * NC + 4 * c),
                     (float4*)(&sB[0][r * NC + c * 4]));
    }
    wait_async_all();
    __syncthreads();

    int cur = 0;
    for (int k0 = 0; k0 < K; k0 += 32) {
        const int nxt = cur ^ 1;
        // ---- issue async stage of next chunk (overlaps with WMMA below) ----
        if (k0 + 32 < K) {
            const int kn = k0 + 32;
            for (int f = tid; f < 128; f += nthr) {
                int r = f >> 3, c = f & 7;
                async_copy16((const float4*)(A + (size_t)(mbase + r) * K + kn + 4 * c),
                             (float4*)(&sA[nxt][r * 32 + c * 4]));
            }
            for (int f = tid; f < 8 * NC; f += nthr) {
                int r = f / NC4, c = f % NC4;
                async_copy16((const float4*)(B + (size_t)(kn + r) * NC + 4 * c),
                             (float4*)(&sB[nxt][r * NC + c * 4]));
            }
        }

        // ---- compute on current buffer ----
#pragma unroll
        for (int kk = 0; kk < 32; kk += 4) {
            // A 16x4 layout: lanes 0-15 hold K=kk,kk+1; lanes 16-31 hold K=kk+2,kk+3
            v2f a, b;
            a[0] = sA[cur][l15 * 32 + kk + 2 * half + 0];
            a[1] = sA[cur][l15 * 32 + kk + 2 * half + 1];
            // B 4x16 layout (mirror): lane's column = l15
            b[0] = sB[cur][(kk + 2 * half + 0) * NC + nbase + l15];
            b[1] = sB[cur][(kk + 2 * half + 1) * NC + nbase + l15];
            acc = __builtin_amdgcn_wmma_f32_16x16x4_f32(
                false, a, false, b, (short)0, acc, false, false);
        }

        wait_async_all();
        __syncthreads();
        cur = nxt;
    }

    // C/D layout: VGPR j -> M = j (lanes 0-15) / j+8 (lanes 16-31), N = l15
#pragma unroll
    for (int j = 0; j < 8; ++j) {
        int row = mbase + j + 8 * half;
        C[(size_t)row * NC + nbase + l15] = acc[j];
    }
}

// ---------------- aggregation ----------------
// self-loop init: agg[i,f] = h[i,f] * dinv[i]^2
__global__ void self_init_k(const float* __restrict__ h,
                            const float* __restrict__ dinv,
                            float* __restrict__ agg, int total) {
    int idx = blockIdx.x * blockDim.x + threadIdx.x;
    if (idx < total) {
        float dv = dinv[idx >> 7];
        agg[idx] = h[idx] * dv * dv;
    }
}

// one wave per edge; lane owns 4 features
__global__ void edge_agg_k(const float* __restrict__ h,
                           float* __restrict__ agg,
                           const int* __restrict__ src,
                           const int* __restrict__ dst,
                           const float* __restrict__ dinv, int e) {
    int eid = blockIdx.x * (blockDim.x >> 5) + (threadIdx.x >> 5);
    if (eid >= e) return;
    int lane = threadIdx.x & 31;
    int s = src[eid], d = dst[eid];
    float coef = dinv[s] * dinv[d];
    const float4 hv = *(const float4*)(h + (size_t)s * HID + lane * 4);
    float* ap = agg + (size_t)d * HID + lane * 4;
    atomicAdd(ap + 0, hv.x * coef);
    atomicAdd(ap + 1, hv.y * coef);
    atomicAdd(ap + 2, hv.z * coef);
    atomicAdd(ap + 3, hv.w * coef);
}

__global__ void bias_relu_k(float* __restrict__ agg, const float* __restrict__ b, int total) {
    int idx = blockIdx.x * blockDim.x + threadIdx.x;
    if (idx < total) {
        float v = agg[idx] + b[idx & (HID - 1)];
        agg[idx] = v > 0.0f ? v : 0.0f;
    }
}

// ---------------- classifier helpers ----------------
__global__ void pack_wc_k(const float* __restrict__ Wc, float* __restrict__ Wp) {
    int idx = blockIdx.x * blockDim.x + threadIdx.x;
    if (idx < HID * 16) {
        int k = idx >> 4, c = idx & 15;
        Wp[idx] = (c < NCLS) ? Wc[k * NCLS + c] : 0.0f;
    }
}

__global__ void out_k(const float* __restrict__ lp, const float* __restrict__ bc,
                      float* __restrict__ out, int total) {
    int idx = blockIdx.x * blockDim.x + threadIdx.x;
    if (idx < total) {
        int i = idx / NCLS, c = idx % NCLS;
        out[idx] = lp[(size_t)i * 16 + c] + bc[c];
    }
}

// ---------------- launch ----------------
extern "C" void kernel_launch(void* const* d_in, const int* in_sizes, int n_in,
                              void* d_out, int out_size, void* d_ws, size_t ws_size,
                              hipStream_t stream) {
    const float* x  = (const float*)d_in[0];
    const int*   ei = (const int*)d_in[1];
    const float* W1 = (const float*)d_in[2];
    const float* b1 = (const float*)d_in[3];
    const float* W2 = (const float*)d_in[4];
    const float* b2 = (const float*)d_in[5];
    const float* Wc = (const float*)d_in[6];
    const float* bc = (const float*)d_in[7];
    float* out = (float*)d_out;

    const int* src = ei;
    const int* dst = ei + E_EDGES;

    float* ws   = (float*)d_ws;
    float* dinv = ws;                                  // N
    float* h    = ws + N_NODES;                        // N*128
    float* agg  = h + (size_t)N_NODES * HID;           // N*128
    float* Wp   = h;                                   // 128*16 (h region dead at that point)
    float* lp   = h + 4096;                            // N*16   (still inside h region)

    const int NF = N_NODES * HID;  // 12.8M

    // normalization: deg = 1 + in-degree; dinv = rsqrt(deg)
    init_deg_k<<<(N_NODES + 255) / 256, 256, 0, stream>>>(dinv, N_NODES);
    count_deg_k<<<(E_EDGES + 255) / 256, 256, 0, stream>>>(dst, dinv, E_EDGES);
    dinv_k<<<(N_NODES + 255) / 256, 256, 0, stream>>>(dinv, N_NODES);

    // layer 1: h = x @ W1 ; agg = Ahat h ; relu(agg + b1)
    gemm16_wmma_k<IN_DIM, HID><<<N_NODES / 16, 256, 0, stream>>>(x, W1, h);
    self_init_k<<<(NF + 255) / 256, 256, 0, stream>>>(h, dinv, agg, NF);
    edge_agg_k<<<(E_EDGES + 7) / 8, 256, 0, stream>>>(h, agg, src, dst, dinv, E_EDGES);
    bias_relu_k<<<(NF + 255) / 256, 256, 0, stream>>>(agg, b1, NF);

    // layer 2: h = agg @ W2 ; agg = Ahat h ; relu(agg + b2)
    gemm16_wmma_k<HID, HID><<<N_NODES / 16, 256, 0, stream>>>(agg, W2, h);
    self_init_k<<<(NF + 255) / 256, 256, 0, stream>>>(h, dinv, agg, NF);
    edge_agg_k<<<(E_EDGES + 7) / 8, 256, 0, stream>>>(h, agg, src, dst, dinv, E_EDGES);
    bias_relu_k<<<(NF + 255) / 256, 256, 0, stream>>>(agg, b2, NF);

    // classifier: logits = agg @ Wc + bc (Wc padded to 16 cols for WMMA)
    pack_wc_k<<<(HID * 16 + 255) / 256, 256, 0, stream>>>(Wc, Wp);
    gemm16_wmma_k<HID, 16><<<N_NODES / 16, 32, 0, stream>>>(agg, Wp, lp);
    out_k<<<(N_NODES * NCLS + 255) / 256, 256, 0, stream>>>(lp, bc, out, N_NODES * NCLS);
}